// GIN_58291296141328
// MI455X (gfx1250) — compile-verified
//
#include <hip/hip_runtime.h>
#include <hip/hip_bf16.h>

typedef __attribute__((ext_vector_type(2))) float v2f;
typedef __attribute__((ext_vector_type(8))) float v8f;

#define N_NODES  50000
#define N_EDGES  640000
#define DIM      128
#define N_GRAPHS 256
#define LATENT   64
#define BN_EPS_F 1e-5f
#define SLOPE_F  0.2f
#define HSTRIDE  130   // padded LDS row stride (even -> keeps v2f 8B alignment)

// ---------------------------------------------------------------- copy x->agg
__global__ void gin_copy_kernel(const float* __restrict__ src,
                                float* __restrict__ dst, int n4) {
    int i = blockIdx.x * blockDim.x + threadIdx.x;
    if (i < n4) ((float4*)dst)[i] = ((const float4*)src)[i];
}

// ----------------------------------------------- agg[dst] += x[src] per edge
__global__ void gin_scatter_kernel(const float* __restrict__ x,
                                   const int* __restrict__ esrc,
                                   const int* __restrict__ edst,
                                   float* __restrict__ agg) {
    int t    = blockIdx.x * blockDim.x + threadIdx.x;
    int e    = t >> 5;
    int lane = t & 31;
    if (e >= N_EDGES) return;
    int s = esrc[e];
    int d = edst[e];
    const float* xs = x + (size_t)s * DIM;
    float*       ad = agg + (size_t)d * DIM;
#pragma unroll
    for (int i = 0; i < 4; ++i) {
        int c = lane + 32 * i;           // wave covers 32 consecutive floats
        atomicAdd(&ad[c], xs[c]);
    }
}

// ---------------- fused MLP: GEMM1 -> leaky -> BN -> GEMM2 -> leaky (f32 WMMA)
__global__ __launch_bounds__(256) void gin_mlp_kernel(
    const float* __restrict__ agg,
    const float* __restrict__ W1, const float* __restrict__ b1,
    const float* __restrict__ bn_g, const float* __restrict__ bn_b,
    const float* __restrict__ bn_m, const float* __restrict__ bn_v,
    const float* __restrict__ W2, const float* __restrict__ b2,
    float* __restrict__ xout)
{
    __shared__ float hshm[8][16 * HSTRIDE];   // per-wave 16x128 h tile (padded)

    const int wave  = threadIdx.x >> 5;
    const int lane  = threadIdx.x & 31;
    const int mrow  = lane & 15;    // M for A, N for B/C/D
    const int khalf = lane >> 4;    // selects K pair {0,1} vs {2,3}
    const int row0  = blockIdx.x * 128 + wave * 16;

    const v8f vzero = {};
    v8f acc[8];
#pragma unroll
    for (int t = 0; t < 8; ++t) acc[t] = vzero;

    int arow = row0 + mrow;
    if (arow >= N_NODES) arow = N_NODES - 1;      // clamp; stores are guarded
    const float* ap = agg + (size_t)arow * DIM;

    // -------- GEMM1: h = agg @ W1^T  (A 16x4 f32, B 4x16 f32, 8 N-tiles)
    for (int kk = 0; kk < DIM; kk += 4) {
        v2f a = *(const v2f*)(ap + kk + 2 * khalf);
#pragma unroll
        for (int t = 0; t < 8; ++t) {
            int col = t * 16 + mrow;
            v2f b = *(const v2f*)(W1 + (size_t)col * DIM + kk + 2 * khalf);
            acc[t] = __builtin_amdgcn_wmma_f32_16x16x4_f32(
                false, a, false, b, (short)0, acc[t], false, false);
        }
    }

    // -------- epilogue 1: +b1, LeakyReLU, BatchNorm(eval) -> LDS
    float* hW = hshm[wave];
#pragma unroll
    for (int t = 0; t < 8; ++t) {
        int col     = t * 16 + mrow;
        float scale = bn_g[col] * rsqrtf(bn_v[col] + BN_EPS_F);
        float shift = bn_b[col] - bn_m[col] * scale;
        float bias  = b1[col];
#pragma unroll
        for (int v = 0; v < 8; ++v) {
            int r   = v + 8 * khalf;                 // D layout row-in-tile
            float h = acc[t][v] + bias;
            h = (h >= 0.f) ? h : SLOPE_F * h;
            h = h * scale + shift;
            hW[r * HSTRIDE + col] = h;
        }
    }
    __syncthreads();

    // -------- GEMM2: x = h @ W2^T, A fragments from LDS
    v8f acc2[8];
#pragma unroll
    for (int t = 0; t < 8; ++t) acc2[t] = vzero;

    const float* hr = hW + mrow * HSTRIDE;
    for (int kk = 0; kk < DIM; kk += 4) {
        v2f a = *(const v2f*)(hr + kk + 2 * khalf);
#pragma unroll
        for (int t = 0; t < 8; ++t) {
            int col = t * 16 + mrow;
            v2f b = *(const v2f*)(W2 + (size_t)col * DIM + kk + 2 * khalf);
            acc2[t] = __builtin_amdgcn_wmma_f32_16x16x4_f32(
                false, a, false, b, (short)0, acc2[t], false, false);
        }
    }

    // -------- epilogue 2: +b2, LeakyReLU -> global
#pragma unroll
    for (int t = 0; t < 8; ++t) {
        int col    = t * 16 + mrow;
        float bias = b2[col];
#pragma unroll
        for (int v = 0; v < 8; ++v) {
            int r   = row0 + v + 8 * khalf;
            float h = acc2[t][v] + bias;
            h = (h >= 0.f) ? h : SLOPE_F * h;
            if (r < N_NODES) xout[(size_t)r * DIM + col] = h;
        }
    }
}

// ------------------------------------------------------------- zero workspace
__global__ void gin_zero_kernel(float* __restrict__ p, int n) {
    int i = blockIdx.x * blockDim.x + threadIdx.x;
    if (i < n) p[i] = 0.f;
}

// ---------------------------------------------------- pooled[batch[n]] += x[n]
__global__ void gin_pool_kernel(const float* __restrict__ x,
                                const int* __restrict__ batch,
                                float* __restrict__ pooled) {
    int t    = blockIdx.x * blockDim.x + threadIdx.x;
    int nid  = t >> 5;
    int lane = t & 31;
    if (nid >= N_NODES) return;
    int g = batch[nid];
    const float* xr = x + (size_t)nid * DIM;
    float*       pr = pooled + (size_t)g * DIM;
#pragma unroll
    for (int i = 0; i < 4; ++i) {
        int c = lane + 32 * i;
        atomicAdd(&pr[c], xr[c]);
    }
}

// --------------------- head: out = BN(pooled) @ fc_W^T + fc_b   (f32 WMMA)
__global__ __launch_bounds__(128) void gin_head_kernel(
    const float* __restrict__ pooled,
    const float* __restrict__ out_g, const float* __restrict__ out_b,
    const float* __restrict__ out_m, const float* __restrict__ out_v,
    const float* __restrict__ fcW, const float* __restrict__ fcb,
    float* __restrict__ out)
{
    const int wave  = threadIdx.x >> 5;
    const int lane  = threadIdx.x & 31;
    const int mrow  = lane & 15;
    const int khalf = lane >> 4;
    const int row0  = blockIdx.x * 64 + wave * 16;

    const v8f vzero = {};
    v8f acc[4];
#pragma unroll
    for (int t = 0; t < 4; ++t) acc[t] = vzero;

    const float* pr = pooled + (size_t)(row0 + mrow) * DIM;

    for (int kk = 0; kk < DIM; kk += 4) {
        int k = kk + 2 * khalf;
        v2f p  = *(const v2f*)(pr + k);
        v2f m  = *(const v2f*)(out_m + k);
        v2f vv = *(const v2f*)(out_v + k);
        v2f g  = *(const v2f*)(out_g + k);
        v2f bb = *(const v2f*)(out_b + k);
        v2f a;
        a.x = (p.x - m.x) * rsqrtf(vv.x + BN_EPS_F) * g.x + bb.x;
        a.y = (p.y - m.y) * rsqrtf(vv.y + BN_EPS_F) * g.y + bb.y;
#pragma unroll
        for (int t = 0; t < 4; ++t) {
            int col = t * 16 + mrow;
            v2f b = *(const v2f*)(fcW + (size_t)col * DIM + k);
            acc[t] = __builtin_amdgcn_wmma_f32_16x16x4_f32(
                false, a, false, b, (short)0, acc[t], false, false);
        }
    }

#pragma unroll
    for (int t = 0; t < 4; ++t) {
        int col    = t * 16 + mrow;
        float bias = fcb[col];
#pragma unroll
        for (int v = 0; v < 8; ++v) {
            int r = row0 + v + 8 * khalf;
            out[(size_t)r * LATENT + col] = acc[t][v] + bias;
        }
    }
}

// ----------------------------------------------------------------- launcher
extern "C" void kernel_launch(void* const* d_in, const int* in_sizes, int n_in,
                              void* d_out, int out_size, void* d_ws, size_t ws_size,
                              hipStream_t stream) {
    const float* x     = (const float*)d_in[0];
    const float* W1    = (const float*)d_in[1];
    const float* b1    = (const float*)d_in[2];
    const float* bn_g  = (const float*)d_in[3];
    const float* bn_b  = (const float*)d_in[4];
    const float* bn_m  = (const float*)d_in[5];
    const float* bn_v  = (const float*)d_in[6];
    const float* W2    = (const float*)d_in[7];
    const float* b2    = (const float*)d_in[8];
    const float* out_g = (const float*)d_in[9];
    const float* out_b = (const float*)d_in[10];
    const float* out_m = (const float*)d_in[11];
    const float* out_v = (const float*)d_in[12];
    const float* fcW   = (const float*)d_in[13];
    const float* fcb   = (const float*)d_in[14];
    const int*   eidx  = (const int*)d_in[15];   // [2, E]
    const int*   batch = (const int*)d_in[16];   // [N]

    float* agg    = (float*)d_ws;                         // [N, D]
    float* xbuf   = agg  + (size_t)N_NODES * DIM;         // [N, D]
    float* pooled = xbuf + (size_t)N_NODES * DIM;         // [G, D]

    const int n4 = N_NODES * DIM / 4;

    for (int l = 0; l < 3; ++l) {
        const float* xin = (l == 0) ? x : xbuf;
        gin_copy_kernel<<<(n4 + 255) / 256, 256, 0, stream>>>(xin, agg, n4);
        gin_scatter_kernel<<<(N_EDGES * 32 + 255) / 256, 256, 0, stream>>>(
            xin, eidx, eidx + N_EDGES, agg);
        gin_mlp_kernel<<<(N_NODES + 127) / 128, 256, 0, stream>>>(
            agg,
            W1 + (size_t)l * DIM * DIM, b1 + l * DIM,
            bn_g + l * DIM, bn_b + l * DIM, bn_m + l * DIM, bn_v + l * DIM,
            W2 + (size_t)l * DIM * DIM, b2 + l * DIM,
            xbuf);
    }

    gin_zero_kernel<<<(N_GRAPHS * DIM + 255) / 256, 256, 0, stream>>>(
        pooled, N_GRAPHS * DIM);
    gin_pool_kernel<<<(N_NODES * 32 + 255) / 256, 256, 0, stream>>>(
        xbuf, batch, pooled);
    gin_head_kernel<<<4, 128, 0, stream>>>(
        pooled, out_g, out_b, out_m, out_v, fcW, fcb, (float*)d_out);
}